// BSQQuantizer_43868795961971
// MI455X (gfx1250) — compile-verified
//
#include <hip/hip_runtime.h>
#include <math.h>

// ---- problem constants (from reference) ----
#define EMBED_DIM  18
#define GROUP_SIZE 9
#define NUM_GROUPS 2
#define NTOK       (32 * 4096)          // 131072 tokens
#define TPB        256                  // threads per block == tokens per block
#define NBLK       (NTOK / TPB)         // 512 blocks
#define QS         0.23570226039551584f // 1/sqrt(18)
#define TWO_QS     (2.0f * QS)
#define ZSTR       19                   // padded LDS stride for zn (odd -> conflict-free)

// ---- output layout (flat concat in reference return order) ----
#define ZQ_N       (NTOK * EMBED_DIM)   // 2359296 floats of zq
#define SCALAR_OFF ZQ_N                 // total, persample, cb_entropy, commit
#define IDX_OFF    (ZQ_N + 4)           // 131072 indices (stored as float, exact < 2^18)

typedef float v2f __attribute__((ext_vector_type(2)));
typedef float v8f __attribute__((ext_vector_type(8)));

// workspace: ws[0..1023] = prob accumulator [2][512]; ws[1024]=commitSum; ws[1025]=entSum
__global__ void bsq_init(float* ws) {
  int i = threadIdx.x + blockIdx.x * blockDim.x;
  if (i < 1026) ws[i] = 0.0f;
}

__global__ __launch_bounds__(TPB) void bsq_main(const float* __restrict__ z,
                                                float* __restrict__ out,
                                                float* __restrict__ ws) {
  __shared__ float zn_lds[TPB * ZSTR];           // normalized z, padded stride
  __shared__ float logz_lds[TPB * NUM_GROUPS];   // per-token per-group log partition
  __shared__ float prob_lds[NUM_GROUPS * 512];   // codebook prob accumulator
  __shared__ float redC[TPB];
  __shared__ float redE[TPB];

  const int tid = threadIdx.x;
  const int t   = blockIdx.x * TPB + tid;        // this thread's token

  for (int i = tid; i < NUM_GROUPS * 512; i += TPB) prob_lds[i] = 0.0f;

  // ---------------- phase 1: per-token scalar work ----------------
  float zv[EMBED_DIM];
  const float* zp = z + (size_t)t * EMBED_DIM;
  float n2 = 0.0f;
#pragma unroll
  for (int c = 0; c < EMBED_DIM; ++c) { zv[c] = zp[c]; n2 += zv[c] * zv[c]; }
  const float inv = 1.0f / fmaxf(sqrtf(n2), 1e-12f);

  float commit = 0.0f, ent = 0.0f;
  float lz0 = 0.0f, lz1 = 0.0f;
  int idx = 0;
#pragma unroll
  for (int c = 0; c < EMBED_DIM; ++c) {
    float zn = zv[c] * inv;
    zn_lds[tid * ZSTR + c] = zn;
    bool pos = zv[c] > 0.0f;
    float zh = pos ? QS : -QS;                       // zq = sign * Q_SCALE (forward)
    out[(size_t)t * EMBED_DIM + c] = zh;
    idx |= (pos ? 1 : 0) << (EMBED_DIM - 1 - c);     // MSB-first basis
    float d = zh - zn;                               // commit: (zq_detached - zn)^2
    commit += d * d;
    float p = 1.0f / (1.0f + __expf(4.0f * QS * zn)); // sigmoid(-4*Q*zn)
    ent -= p * __logf(p + 1e-8f) + (1.0f - p) * __logf(1.0f - p + 1e-8f);
    // log(e^a + e^-a) with a = 2*Q*zn, stable form
    float a = fabsf(TWO_QS * zn);
    float lzc = a + __logf(1.0f + __expf(-2.0f * a));
    if (c < GROUP_SIZE) lz0 += lzc; else lz1 += lzc;
  }
  out[IDX_OFF + t] = (float)idx;
  logz_lds[tid * 2 + 0] = lz0;
  logz_lds[tid * 2 + 1] = lz1;
  __syncthreads();

  // ---------------- phase 2: WMMA logits + factorized softmax ----------------
  // D = A(16 tok x 12) * B(12 x 16 codes), K=9 padded to 12 -> 3 chained f32 WMMAs.
  // f32 16x16x4 operand layout (ISA 7.12.2): lanes 0-15 carry M/N = lane,
  // lanes 16-31 carry the K+2 slice; vgpr e in {0,1} is K = 4*s + e (+2 for hi lanes).
  const int wave = tid >> 5;
  const int lane = tid & 31;
  const int ln   = lane & 15;
  const int khi  = (lane >= 16) ? 2 : 0;
  const int mhi  = (lane >= 16) ? 8 : 0;

  for (int tt = 0; tt < 2; ++tt) {                 // 8 waves x 2 tiles = 256 tokens
    const int tb = (wave * 2 + tt) * 16;
    const int mA = tb + ln;
#pragma unroll
    for (int g = 0; g < NUM_GROUPS; ++g) {
      v2f a[3];
#pragma unroll
      for (int s = 0; s < 3; ++s)
#pragma unroll
        for (int e = 0; e < 2; ++e) {
          int K = 4 * s + e + khi;
          a[s][e] = (K < GROUP_SIZE) ? zn_lds[mA * ZSTR + GROUP_SIZE * g + K] : 0.0f;
        }
      float lzv[8];
#pragma unroll
      for (int r = 0; r < 8; ++r) lzv[r] = logz_lds[(tb + r + mhi) * 2 + g];

      for (int ct = 0; ct < 32; ++ct) {            // 32 tiles of 16 codes = 512
        const int code = ct * 16 + ln;
        v2f b[3];
#pragma unroll
        for (int s = 0; s < 3; ++s)
#pragma unroll
          for (int e = 0; e < 2; ++e) {
            int K = 4 * s + e + khi;
            // fold logits scale 2*Q_SCALE into B; codebook bit c=K is MSB-first
            b[s][e] = (K < GROUP_SIZE)
                        ? (((code >> (GROUP_SIZE - 1 - K)) & 1) ? TWO_QS : -TWO_QS)
                        : 0.0f;
          }
        v8f acc = {0.f, 0.f, 0.f, 0.f, 0.f, 0.f, 0.f, 0.f};
        acc = __builtin_amdgcn_wmma_f32_16x16x4_f32(false, a[0], false, b[0], (short)0, acc, false, false);
        acc = __builtin_amdgcn_wmma_f32_16x16x4_f32(false, a[1], false, b[1], (short)0, acc, false, false);
        acc = __builtin_amdgcn_wmma_f32_16x16x4_f32(false, a[2], false, b[2], (short)0, acc, false, false);
        // prob = exp(logit - logZ); sum over this lane's 8 tokens, accumulate per code
        float s8 = 0.0f;
#pragma unroll
        for (int r = 0; r < 8; ++r) s8 += __expf(acc[r] - lzv[r]);
        atomicAdd(&prob_lds[g * 512 + ct * 16 + ln], s8);
      }
    }
  }
  __syncthreads();

  // ---------------- reductions to global workspace ----------------
  redC[tid] = commit;
  redE[tid] = ent;
  __syncthreads();
  for (int s = TPB / 2; s > 0; s >>= 1) {
    if (tid < s) { redC[tid] += redC[tid + s]; redE[tid] += redE[tid + s]; }
    __syncthreads();
  }
  if (tid == 0) {
    atomicAdd(&ws[1024], redC[0]);
    atomicAdd(&ws[1025], redE[0]);
  }
  for (int i = tid; i < NUM_GROUPS * 512; i += TPB)
    atomicAdd(&ws[i], prob_lds[i]);
}

__global__ __launch_bounds__(256) void bsq_final(const float* __restrict__ ws,
                                                 float* __restrict__ out) {
  __shared__ float red[256];
  const int tid = threadIdx.x;
  float part = 0.0f;
  for (int i = tid; i < 1024; i += 256) {
    float avg = ws[i] * (1.0f / (float)NTOK);     // mean over B,L
    part -= avg * __logf(avg + 1e-8f);
  }
  red[tid] = part;
  __syncthreads();
  for (int s = 128; s > 0; s >>= 1) {
    if (tid < s) red[tid] += red[tid + s];
    __syncthreads();
  }
  if (tid == 0) {
    float cbe    = red[0];
    float commit = ws[1024] * (1.0f / (float)NTOK);
    float pers   = ws[1025] * (1.0f / (float)NTOK);
    // BETA=GAMMA0=GAMMA=ZETA=INV_TEMP=1
    out[SCALAR_OFF + 0] = commit + pers - cbe;  // total_loss
    out[SCALAR_OFF + 1] = pers;                 // persample_entropy
    out[SCALAR_OFF + 2] = cbe;                  // cb_entropy
    out[SCALAR_OFF + 3] = commit;               // commit_loss
  }
}

extern "C" void kernel_launch(void* const* d_in, const int* in_sizes, int n_in,
                              void* d_out, int out_size, void* d_ws, size_t ws_size,
                              hipStream_t stream) {
  (void)in_sizes; (void)n_in; (void)out_size; (void)ws_size;
  const float* z  = (const float*)d_in[0];
  float*       out = (float*)d_out;
  float*       ws  = (float*)d_ws;

  bsq_init<<<dim3(8), dim3(256), 0, stream>>>(ws);
  bsq_main<<<dim3(NBLK), dim3(TPB), 0, stream>>>(z, out, ws);
  bsq_final<<<dim3(1), dim3(256), 0, stream>>>(ws, out);
}